// Inter_channel_27084063769110
// MI455X (gfx1250) — compile-verified
//
#include <hip/hip_runtime.h>
#include <hip/hip_bf16.h>

// Problem constants (from reference): B=2, C=512, H=256, W=256
#define CC   512
#define HH   256
#define WW   256
#define EPSV 1e-5f

typedef float v2f __attribute__((ext_vector_type(2)));
typedef float v4f __attribute__((ext_vector_type(4)));
typedef float v8f __attribute__((ext_vector_type(8)));

// ---------------------------------------------------------------------------
// Kernel 1: per (b,c) plane, single pass over x:
//   conv1[w] = sum_h x[h,w]*w1[h]   (stored [C,B,W])
//   conv2[h] = sum_w x[h,w]*w2[w]   (stored [C,B,H])
// 256 threads: thread t -> row group j=t>>6 (4 rows/iter), col group q=t&63
// (float4 = 4 columns). Row reductions via wave shuffles, per-wave LDS slots.
// ---------------------------------------------------------------------------
__global__ __launch_bounds__(256) void k1_conv(
    const float* __restrict__ x, const float* __restrict__ w1,
    const float* __restrict__ w2, float* __restrict__ conv1,
    float* __restrict__ conv2) {
  __shared__ float  w1s[HH];
  __shared__ float  part[8][64];   // [wave][iter]
  __shared__ float4 c1part[256];

  const int t    = threadIdx.x;
  const int p    = blockIdx.x;          // b*C + c (matches x plane order)
  const int b    = p >> 9;
  const int c    = p & (CC - 1);
  const int q    = t & 63;              // column group (4 cols)
  const int j    = t >> 6;              // row offset within 4-row group
  const int wave = t >> 5;
  const int lane = t & 31;

  const float* __restrict__ X = x + (size_t)p * (HH * WW);
  w1s[t] = w1[c * HH + t];
  __syncthreads();

  const float4 w2v = ((const float4*)(w2 + c * WW))[q];
  float4 acc1 = make_float4(0.f, 0.f, 0.f, 0.f);

  for (int i = 0; i < 64; ++i) {
    const int r = 4 * i + j;
    const float4 xv = ((const float4*)(X + (size_t)r * WW))[q];
    const float w1r = w1s[r];
    acc1.x += xv.x * w1r; acc1.y += xv.y * w1r;
    acc1.z += xv.z * w1r; acc1.w += xv.w * w1r;
    float s = xv.x * w2v.x + xv.y * w2v.y + xv.z * w2v.z + xv.w * w2v.w;
    s += __shfl_xor(s, 16, 32);
    s += __shfl_xor(s,  8, 32);
    s += __shfl_xor(s,  4, 32);
    s += __shfl_xor(s,  2, 32);
    s += __shfl_xor(s,  1, 32);
    if (lane == 0) part[wave][i] = s;   // wave owns its slot; no barrier needed
  }
  c1part[t] = acc1;
  __syncthreads();

  // conv2[r]: row r was handled in iter r>>2 by waves 2*(r&3), 2*(r&3)+1
  {
    const int r = t;
    const float v = part[2 * (r & 3)][r >> 2] + part[2 * (r & 3) + 1][r >> 2];
    conv2[((size_t)c * 2 + b) * HH + r] = v;
  }
  // conv1: fold the 4 row-group accumulators per column group
  if (t < 64) {
    const float4 a0 = c1part[t], a1 = c1part[t + 64];
    const float4 a2 = c1part[t + 128], a3 = c1part[t + 192];
    float4 o = make_float4(a0.x + a1.x + a2.x + a3.x, a0.y + a1.y + a2.y + a3.y,
                           a0.z + a1.z + a2.z + a3.z, a0.w + a1.w + a2.w + a3.w);
    ((float4*)(conv1 + ((size_t)c * 2 + b) * WW))[t] = o;
  }
}

// ---------------------------------------------------------------------------
// Block-wide (256 thread) sum reduction helper
// ---------------------------------------------------------------------------
__device__ __forceinline__ float block_sum_256(float v, float* sred) {
  v += __shfl_xor(v, 16, 32);
  v += __shfl_xor(v,  8, 32);
  v += __shfl_xor(v,  4, 32);
  v += __shfl_xor(v,  2, 32);
  v += __shfl_xor(v,  1, 32);
  if ((threadIdx.x & 31) == 0) sred[threadIdx.x >> 5] = v;
  __syncthreads();
  float r = sred[0] + sred[1] + sred[2] + sred[3] +
            sred[4] + sred[5] + sred[6] + sred[7];
  __syncthreads();
  return r;
}

// ---------------------------------------------------------------------------
// Kernel 2: per channel c: BN (biased stats over the 512 values spanning B,W)
// + ReLU for both branches, then per-batch dot products.
// Writes num_flat[2c+b] -- exactly the reference's [c,b]->view(b,c) scramble.
// ---------------------------------------------------------------------------
__global__ __launch_bounds__(256) void k2_bn_dot(
    const float* __restrict__ conv1, const float* __restrict__ conv2,
    const float* __restrict__ g1, const float* __restrict__ bb1,
    const float* __restrict__ g2, const float* __restrict__ bb2,
    float* __restrict__ numbuf) {
  __shared__ float sred[8];
  const int c = blockIdx.x, t = threadIdx.x;
  const float* v1 = conv1 + (size_t)c * 512;   // [b=0 w:256][b=1 w:256]
  const float* v2 = conv2 + (size_t)c * 512;

  const float a0 = v1[t], a1 = v1[t + 256];
  const float e0 = v2[t], e1 = v2[t + 256];

  const float S1 = block_sum_256(a0 + a1, sred);
  const float Q1 = block_sum_256(a0 * a0 + a1 * a1, sred);
  const float S2 = block_sum_256(e0 + e1, sred);
  const float Q2 = block_sum_256(e0 * e0 + e1 * e1, sred);

  const float m1 = S1 * (1.f / 512.f);
  const float m2 = S2 * (1.f / 512.f);
  const float i1 = __frsqrt_rn(fmaxf(Q1 * (1.f / 512.f) - m1 * m1, 0.f) + EPSV);
  const float i2 = __frsqrt_rn(fmaxf(Q2 * (1.f / 512.f) - m2 * m2, 0.f) + EPSV);
  const float G1 = g1[c], B1 = bb1[c], G2 = g2[c], B2 = bb2[c];

  const float x10 = fmaxf((a0 - m1) * i1 * G1 + B1, 0.f);
  const float x11 = fmaxf((a1 - m1) * i1 * G1 + B1, 0.f);
  const float x20 = fmaxf((e0 - m2) * i2 * G2 + B2, 0.f);
  const float x21 = fmaxf((e1 - m2) * i2 * G2 + B2, 0.f);

  const float d0 = block_sum_256(x10 * x20, sred);  // batch 0
  const float d1 = block_sum_256(x11 * x21, sred);  // batch 1
  if (t == 0) {
    numbuf[2 * c + 0] = d0;
    numbuf[2 * c + 1] = d1;
  }
}

// ---------------------------------------------------------------------------
// Kernel 3: MLP via V_WMMA_F32_16X16X4_F32 (exact f32 path).
// A = num zero-padded to [16,512] IN LDS (rows 2..15 = 0) so all fragment
// loads are uniform, branch-free ds_load_b64 (EXEC stays all-ones for WMMA).
// y1 = A*lw1^T + lb1 -> LDS [16,1024];  y2 = y1*lw2^T + lb2, rows 0..1 kept.
// Bias-contaminated padded rows never feed the real rows (row-independent).
// A-frag (16x4 f32): lane<16 holds K={k,k+1} of row M=lane; lane>=16 K={k+2,k+3}.
// B-frag mirrors transposed; C/D v8f: D[M = r + 8*(lane>=16)][N = n0+(lane&15)].
// ---------------------------------------------------------------------------
__global__ __launch_bounds__(256) void k3_mlp(
    const float* __restrict__ numbuf, const float* __restrict__ lw1,
    const float* __restrict__ lb1, const float* __restrict__ lw2,
    const float* __restrict__ lb2, float* __restrict__ y2buf) {
  extern __shared__ float smem[];
  float* As  = smem;              // [16][512] zero-padded A (32 KiB)
  float* y1s = smem + 16 * 512;   // [16][1024]              (64 KiB)

  const int t = threadIdx.x;
  for (int i = t; i < 16 * 512; i += 256)
    As[i] = (i < 1024) ? numbuf[i] : 0.f;   // rows 0..1 = num, rest zero
  __syncthreads();

  const int wave  = t >> 5;
  const int lane  = t & 31;
  const int m16   = lane & 15;
  const int halfk = lane >> 4;

  // ---- Stage 1: y1[16][1024] = A(16x512) * lw1^T + lb1 ----
  for (int nt = wave; nt < 64; nt += 8) {
    const int n0 = nt * 16;
    const int n  = n0 + m16;
    const float bias = lb1[n];
    v8f acc;
#pragma unroll
    for (int r = 0; r < 8; ++r) acc[r] = bias;
    const float* __restrict__ Brow = lw1 + (size_t)n * 512;
    const float* __restrict__ Arow = As + m16 * 512;
    for (int kt = 0; kt < 128; ++kt) {
      const int k = 4 * kt + 2 * halfk;          // even -> 8B aligned
      const v2f a    = *(const v2f*)(Arow + k);  // one ds_load_b64
      const v2f bvec = *(const v2f*)(Brow + k);  // one global_load_b64
      acc = __builtin_amdgcn_wmma_f32_16x16x4_f32(
          false, a, false, bvec, (short)0, acc, false, false);
    }
#pragma unroll
    for (int r = 0; r < 8; ++r)
      y1s[(r + 8 * halfk) * 1024 + n0 + m16] = acc[r];
  }
  __syncthreads();

  // ---- Stage 2: y2[16][512] = y1(16x1024) * lw2^T + lb2; keep rows 0..1 ----
  for (int nt = wave; nt < 32; nt += 8) {
    const int n0 = nt * 16;
    const int n  = n0 + m16;
    const float bias = lb2[n];
    v8f acc;
#pragma unroll
    for (int r = 0; r < 8; ++r) acc[r] = bias;
    const float* __restrict__ Brow = lw2 + (size_t)n * 1024;
    const float* __restrict__ Arow = y1s + m16 * 1024;
    for (int kt = 0; kt < 256; ++kt) {
      const int k = 4 * kt + 2 * halfk;
      const v2f a    = *(const v2f*)(Arow + k);
      const v2f bvec = *(const v2f*)(Brow + k);
      acc = __builtin_amdgcn_wmma_f32_16x16x4_f32(
          false, a, false, bvec, (short)0, acc, false, false);
    }
    if (halfk == 0) {                 // rows m=0 (vgpr0), m=1 (vgpr1)
      y2buf[0 * 512 + n0 + m16] = acc[0];
      y2buf[1 * 512 + n0 + m16] = acc[1];
    }
  }
}

// ---------------------------------------------------------------------------
// Kernel 4: row softmax over [2,512]
// ---------------------------------------------------------------------------
__global__ __launch_bounds__(256) void k4_softmax(
    const float* __restrict__ y2buf, float* __restrict__ smbuf) {
  __shared__ float sred[8];
  const int b = blockIdx.x, t = threadIdx.x;
  const float* row = y2buf + (size_t)b * 512;
  const float v0 = row[t], v1 = row[t + 256];

  float m = fmaxf(v0, v1);
  m = fmaxf(m, __shfl_xor(m, 16, 32));
  m = fmaxf(m, __shfl_xor(m,  8, 32));
  m = fmaxf(m, __shfl_xor(m,  4, 32));
  m = fmaxf(m, __shfl_xor(m,  2, 32));
  m = fmaxf(m, __shfl_xor(m,  1, 32));
  if ((t & 31) == 0) sred[t >> 5] = m;
  __syncthreads();
  m = fmaxf(fmaxf(fmaxf(sred[0], sred[1]), fmaxf(sred[2], sred[3])),
            fmaxf(fmaxf(sred[4], sred[5]), fmaxf(sred[6], sred[7])));
  __syncthreads();

  const float e0 = __expf(v0 - m), e1 = __expf(v1 - m);
  const float S  = block_sum_256(e0 + e1, sred);
  const float inv = 1.f / S;
  smbuf[b * 512 + t]       = e0 * inv;
  smbuf[b * 512 + t + 256] = e1 * inv;
}

// ---------------------------------------------------------------------------
// Kernel 5: out = x * scale[plane]  (dominant 512 MiB traffic; B128 vectorized,
// plane index is block-uniform so the scale load scalarizes). Output is never
// re-read and x (256 MiB) exceeds the 192 MB L2 -> non-temporal hints, using
// native ext_vector types (the builtin rejects HIP_vector_type structs).
// ---------------------------------------------------------------------------
__global__ __launch_bounds__(256) void k5_scale(
    const float* __restrict__ x, const float* __restrict__ smbuf,
    float* __restrict__ out) {
  const size_t i4 = (size_t)blockIdx.x * 256 + threadIdx.x;  // float4 index
  const int plane = (int)(i4 >> 14);         // 16384 float4 per 256x256 plane
  const float s = smbuf[plane];              // plane = b*512 + c
  v4f v = __builtin_nontemporal_load(((const v4f*)x) + i4);
  v *= s;
  __builtin_nontemporal_store(v, ((v4f*)out) + i4);
}

// ---------------------------------------------------------------------------
extern "C" void kernel_launch(void* const* d_in, const int* in_sizes, int n_in,
                              void* d_out, int out_size, void* d_ws, size_t ws_size,
                              hipStream_t stream) {
  const float* x   = (const float*)d_in[0];
  const float* w1  = (const float*)d_in[1];
  const float* w2  = (const float*)d_in[2];
  const float* g1  = (const float*)d_in[3];
  const float* b1  = (const float*)d_in[4];
  const float* g2  = (const float*)d_in[5];
  const float* b2  = (const float*)d_in[6];
  const float* lw1 = (const float*)d_in[7];
  const float* lb1 = (const float*)d_in[8];
  const float* lw2 = (const float*)d_in[9];
  const float* lb2 = (const float*)d_in[10];
  float* out = (float*)d_out;

  // Workspace layout (floats)
  float* conv1  = (float*)d_ws;                   // [C,B,W]  = 262144
  float* conv2  = conv1 + 262144;                 // [C,B,H]  = 262144
  float* numbuf = conv2 + 262144;                 // [2,512]  (scrambled view)
  float* y2buf  = numbuf + 1024;                  // [2,512]
  float* smbuf  = y2buf + 1024;                   // [2,512]

  k1_conv<<<2 * CC, 256, 0, stream>>>(x, w1, w2, conv1, conv2);
  k2_bn_dot<<<CC, 256, 0, stream>>>(conv1, conv2, g1, b1, g2, b2, numbuf);
  k3_mlp<<<1, 256, (16 * 512 + 16 * 1024) * sizeof(float), stream>>>(
      numbuf, lw1, lb1, lw2, lb2, y2buf);
  k4_softmax<<<2, 256, 0, stream>>>(y2buf, smbuf);
  k5_scale<<<(2 * CC * HH * WW / 4) / 256, 256, 0, stream>>>(x, smbuf, out);
}